// JacobiMachine_88553635709529
// MI455X (gfx1250) — compile-verified
//
#include <hip/hip_runtime.h>
#include <math.h>

// Jacobi 5-point stencil, 4096^2 f32, 100 steps.
// Memory-bound (0.6 FLOP/byte): the whole game is data movement.
//  - ping-pong buffers (2 x 67MB) both fit in MI455X's 192MB L2 -> default RT hints
//  - CDNA5 async global->LDS DMA (ASYNCcnt) stages halo tiles, B128 for the bulk
//  - all hot paths 128-bit wide: async b128 in, ds_load_b128 compute, b128 stores
//  - wave32: each wave stores 32 consecutive float4s = 512B contiguous

#define GN 4096
#define NSTEPS 100

#define TX 128                 // outputs per block, x
#define TY 32                  // outputs per block, y
#define PITCH 136              // LDS row pitch in floats (16B-aligned quads)
// LDS layout: staged col index = (global col - (tc0-1)) + 3
//   -> left halo (tc0-1) at col 3, interior tc0..tc0+127 at cols 4..131 (16B aligned),
//      right halo (tc0+128) at col 132. Staged rows 0..33 = grid rows tr0-1 .. tr0+32.
#define S_FLOATS (34 * PITCH)  // 4624 floats = 18.5 KB LDS

__global__ __launch_bounds__(256)
void jacobi_init_kernel(const float* __restrict__ X,
                        const float* __restrict__ Y,
                        float* __restrict__ U0) {
    size_t k = (size_t)blockIdx.x * blockDim.x + threadIdx.x;  // one float4/thread
    const float4* x4 = (const float4*)X;
    const float4* y4 = (const float4*)Y;
    float4*       u4 = (float4*)U0;
    float4 xv = x4[k];
    float4 yv = y4[k];
    float4 o;
    o.x = expf(-50.0f * ((xv.x - 0.5f) * (xv.x - 0.5f) + (yv.x - 0.5f) * (yv.x - 0.5f)));
    o.y = expf(-50.0f * ((xv.y - 0.5f) * (xv.y - 0.5f) + (yv.y - 0.5f) * (yv.y - 0.5f)));
    o.z = expf(-50.0f * ((xv.z - 0.5f) * (xv.z - 0.5f) + (yv.z - 0.5f) * (yv.z - 0.5f)));
    o.w = expf(-50.0f * ((xv.w - 0.5f) * (xv.w - 0.5f) + (yv.w - 0.5f) * (yv.w - 0.5f)));
    u4[k] = o;
}

__global__ __launch_bounds__(256)
void jacobi_step_kernel(const float* __restrict__ src,
                        float* __restrict__ dst) {
    __shared__ float tile[S_FLOATS];

    const int tc0 = blockIdx.x * TX;   // tile origin col (multiple of 128)
    const int tr0 = blockIdx.y * TY;   // tile origin row
    const int t   = threadIdx.x;

    // generic shared pointer: low 32 bits == LDS byte address
    unsigned ldsBase = (unsigned)(size_t)(&tile[0]);

    // ---- Phase A: bulk (34 rows x 32 quads) via async B128 DMA, 16B/lane ----
    // 34*32 = 1088 quad-transfers; 4 full wavefront passes + one partial (t<64).
#pragma unroll
    for (int i = 0; i < 5; ++i) {
        int m = t + i * 256;
        if (m < 34 * 32) {
            int row = m >> 5;                 // staged row 0..33
            int qj  = m & 31;                 // quad within row
            int gr  = tr0 - 1 + row;          // grid row (clamp: feeds only zeroed edges)
            gr = gr < 0 ? 0 : (gr > GN - 1 ? GN - 1 : gr);
            unsigned long long gaddr = (unsigned long long)(size_t)
                (src + (size_t)gr * GN + (tc0 + (qj << 2)));            // 16B aligned
            unsigned laddr = ldsBase + (unsigned)((row * PITCH + 4 + (qj << 2)) * 4);
            asm volatile("global_load_async_to_lds_b128 %0, %1, off"
                         :: "v"(laddr), "v"(gaddr) : "memory");
        }
    }
    // ---- Phase B: halo columns (34 rows x {left,right}) via async B32 ----
    if (t < 68) {
        int row  = t >> 1;
        int side = t & 1;
        int gr = tr0 - 1 + row;
        gr = gr < 0 ? 0 : (gr > GN - 1 ? GN - 1 : gr);
        int gc = side ? (tc0 + TX) : (tc0 - 1);
        gc = gc < 0 ? 0 : (gc > GN - 1 ? GN - 1 : gc);
        int lcol = side ? (4 + TX) : 3;       // 132 or 3
        unsigned long long gaddr = (unsigned long long)(size_t)
            (src + (size_t)gr * GN + gc);
        unsigned laddr = ldsBase + (unsigned)((row * PITCH + lcol) * 4);
        asm volatile("global_load_async_to_lds_b32 %0, %1, off"
                     :: "v"(laddr), "v"(gaddr) : "memory");
    }
    // wait for this wave's async DMA into LDS, then sync the workgroup
    asm volatile("s_wait_asynccnt 0x0" ::: "memory");
    __syncthreads();

    // ---- compute: 4 outputs per thread-iteration, all LDS reads 128-bit ----
    const float4* tile4 = (const float4*)tile;    // float4 row pitch = 34
#pragma unroll
    for (int i = 0; i < 4; ++i) {
        int q  = t + i * 256;        // quad id 0..1023
        int ly = q >> 5;             // output row in tile 0..31 (staged row ly+1)
        int qx = q & 31;             // quad within row
        int lx = qx << 2;            // output col in tile 0..124

        float4 up  = tile4[(ly    ) * 34 + 1 + qx];   // staged row ly   = grid gr-1
        float4 mid = tile4[(ly + 1) * 34 + 1 + qx];   // staged row ly+1 = grid gr
        float4 dn  = tile4[(ly + 2) * 34 + 1 + qx];   // staged row ly+2 = grid gr+1
        float  l   = tile[(ly + 1) * PITCH + 3 + lx]; // grid col gc-1
        float  r   = tile[(ly + 1) * PITCH + 8 + lx]; // grid col gc+4

        float v0 = 0.25f * ((up.x + dn.x) + (l     + mid.y));
        float v1 = 0.25f * ((up.y + dn.y) + (mid.x + mid.z));
        float v2 = 0.25f * ((up.z + dn.z) + (mid.y + mid.w));
        float v3 = 0.25f * ((up.w + dn.w) + (mid.z + r    ));

        int gr = tr0 + ly;
        int gc = tc0 + lx;
        // zero Dirichlet boundary (jnp.pad(interior, 1))
        if (gr == 0 || gr == GN - 1) { v0 = 0.f; v1 = 0.f; v2 = 0.f; v3 = 0.f; }
        if (gc == 0)           v0 = 0.f;
        if (gc + 3 == GN - 1)  v3 = 0.f;

        float4 o; o.x = v0; o.y = v1; o.z = v2; o.w = v3;
        *(float4*)(dst + (size_t)gr * GN + gc) = o;   // b128, 512B per wave row
    }
}

extern "C" void kernel_launch(void* const* d_in, const int* in_sizes, int n_in,
                              void* d_out, int out_size, void* d_ws, size_t ws_size,
                              hipStream_t stream) {
    const float* X = (const float*)d_in[0];
    const float* Y = (const float*)d_in[1];

    // u_k lives in buffer (k mod 2): buf0 = d_out, buf1 = d_ws (needs 64 MiB ws).
    // u_100 -> buf0 = d_out; every element of d_out rewritten each call.
    float* buf0 = (float*)d_out;
    float* buf1 = (float*)d_ws;

    // init: x0 = exp(-50((X-.5)^2 + (Y-.5)^2)), full grid incl. boundary
    {
        size_t n4 = (size_t)GN * GN / 4;             // 4,194,304 float4s
        dim3 grid((unsigned)(n4 / 256)), block(256);
        jacobi_init_kernel<<<grid, block, 0, stream>>>(X, Y, buf0);
    }

    dim3 grid(GN / TX, GN / TY);                     // 32 x 128 tiles
    dim3 block(256);                                  // 8 wave32s
    for (int s = 0; s < NSTEPS; ++s) {
        const float* sp = (s & 1) ? buf1 : buf0;
        float*       dp = (s & 1) ? buf0 : buf1;
        jacobi_step_kernel<<<grid, block, 0, stream>>>(sp, dp);
    }
}